// centercompute_38027640439207
// MI455X (gfx1250) — compile-verified
//
#include <hip/hip_runtime.h>
#include <hip/hip_bf16.h>

// seg_sum = onehot(labels)^T [4 x N]  x  features [N x 256], via
// V_WMMA_F32_16X16X4_F32 (pure-f32 WMMA, no precision loss vs reference).
//
// Layouts (CDNA5 ISA 7.12.2, wave32):
//   A (16x4 f32, 2 VGPRs): lanes 0-15 -> M=0..15, lanes 16-31 -> M=0..15;
//       VGPR0 holds K=0 (lanes 0-15) / K=2 (lanes 16-31); VGPR1 holds K=1 / K=3.
//   B (4x16 f32, 2 VGPRs): mirrored: VGPR0 = row K0 / K2, VGPR1 = row K1 / K3,
//       N = lane & 15.
//   C (16x16 f32, 8 VGPRs): VGPR r, lanes 0-15 -> M=r, N=lane. Classes 0..3
//       therefore live in c[0..3], lanes 0-15 -> coalesced atomic flush.

typedef __attribute__((ext_vector_type(2))) float v2f;
typedef __attribute__((ext_vector_type(8))) float v8f;

#define DIMS 256
#define TILE 16
#define BLOCK_THREADS 512           // 16 waves; wave w owns column tile w*16
#define ROWS_PER_BLOCK 512

__global__ __launch_bounds__(BLOCK_THREADS)
void seg_sum_wmma_kernel(const float* __restrict__ features,
                         const int* __restrict__ labels,
                         float* __restrict__ seg_sum,   // [4 x 256] accumulators
                         int N) {
    const int  lane    = threadIdx.x & 31;
    const int  wave    = threadIdx.x >> 5;
    const int  colBase = wave * TILE;
    const int  m       = lane & 15;          // A-matrix row (class id) / B col
    const bool hi      = (lane >= 16);       // lane half -> K pair {2,3} vs {0,1}
    const int  kBase   = hi ? 2 : 0;

    const int rowStart = blockIdx.x * ROWS_PER_BLOCK;

    v8f c = {0.f, 0.f, 0.f, 0.f, 0.f, 0.f, 0.f, 0.f};

    if (rowStart + ROWS_PER_BLOCK <= N) {
        // ---- Fast path: whole block in range, no guards. Uniform branch, so
        // EXEC is all-ones at every WMMA. 16 rows (1KB/wave) per iteration.
        const float* fp = features + (long)(rowStart + kBase) * DIMS + colBase + m;
        for (int r = rowStart; r < rowStart + ROWS_PER_BLOCK; r += 16) {
            // Prefetch ~2 chunks ahead (speculative: dropped if OOB).
            __builtin_prefetch(fp + 32 * DIMS, 0, 1);
#pragma unroll
            for (int u = 0; u < 4; ++u) {
                // Wave-uniform address -> s_load (SMEM path), 4 labels at once.
                const int4 lab = *(const int4*)(labels + r + 4 * u);
                const int l0 = hi ? lab.z : lab.x;
                const int l1 = hi ? lab.w : lab.y;
                v2f a, b;
                a.x = (l0 == m) ? 1.0f : 0.0f;   // one-hot A tile
                a.y = (l1 == m) ? 1.0f : 0.0f;
                b.x = fp[0];                      // row r+4u+kBase
                b.y = fp[DIMS];                   // row r+4u+kBase+1
                fp += 4 * DIMS;
                c = __builtin_amdgcn_wmma_f32_16x16x4_f32(
                        false, a, false, b, (short)0, c, false, false);
            }
        }
    } else {
        // ---- Tail path (last block only): guarded, clamped addresses.
        const int rowEnd = N;
        for (int r = rowStart; r < rowEnd; r += 4) {
            const int r0 = r + kBase;
            const int r1 = r0 + 1;
            const int s0 = min(r0, N - 1);
            const int s1 = min(r1, N - 1);
            const int lab0 = labels[s0];
            const int lab1 = labels[s1];
            v2f a, b;
            a.x = (r0 < N && lab0 == m) ? 1.0f : 0.0f;
            a.y = (r1 < N && lab1 == m) ? 1.0f : 0.0f;
            b.x = features[(long)s0 * DIMS + colBase + m];
            b.y = features[(long)s1 * DIMS + colBase + m];
            c = __builtin_amdgcn_wmma_f32_16x16x4_f32(
                    false, a, false, b, (short)0, c, false, false);
        }
    }

    // Flush: classes 0..3 are C VGPRs 0..3, lanes 0..15.
    if (lane < 16) {
#pragma unroll
        for (int cls = 0; cls < 4; ++cls) {
            atomicAdd(&seg_sum[cls * DIMS + colBase + lane], c[cls]);
        }
    }
}

__global__ void zero_ws_kernel(float* __restrict__ ws, int n) {
    int i = blockIdx.x * blockDim.x + threadIdx.x;
    if (i < n) ws[i] = 0.0f;
}

__global__ __launch_bounds__(256)
void count_labels_kernel(const int* __restrict__ labels,
                         float* __restrict__ counts,   // [4]
                         int N) {
    __shared__ float scnt[4];
    if (threadIdx.x < 4) scnt[threadIdx.x] = 0.0f;
    __syncthreads();

    float c0 = 0.f, c1 = 0.f, c2 = 0.f, c3 = 0.f;
    const int stride = gridDim.x * blockDim.x;
    for (int i = blockIdx.x * blockDim.x + threadIdx.x; i < N; i += stride) {
        const int l = labels[i];
        c0 += (l == 0) ? 1.0f : 0.0f;
        c1 += (l == 1) ? 1.0f : 0.0f;
        c2 += (l == 2) ? 1.0f : 0.0f;
        c3 += (l == 3) ? 1.0f : 0.0f;
    }
    atomicAdd(&scnt[0], c0);
    atomicAdd(&scnt[1], c1);
    atomicAdd(&scnt[2], c2);
    atomicAdd(&scnt[3], c3);
    __syncthreads();
    if (threadIdx.x < 4) atomicAdd(&counts[threadIdx.x], scnt[threadIdx.x]);
}

__global__ __launch_bounds__(256)
void finalize_kernel(const float* __restrict__ ws,   // [0..1023]=seg_sum, [1024..1027]=counts
                     float* __restrict__ out) {      // [0..1023]=centers, [1024..1027]=target
    __shared__ float red[256];
    const int t = threadIdx.x;

#pragma unroll
    for (int cls = 0; cls < 4; ++cls) {
        const float cnt = ws[4 * DIMS + cls];
        const float v = ws[cls * DIMS + t] / cnt;    // per-class mean
        red[t] = v * v;
        __syncthreads();
        for (int s = 128; s > 0; s >>= 1) {
            if (t < s) red[t] += red[t + s];
            __syncthreads();
        }
        const float norm = sqrtf(red[0]);
        __syncthreads();
        out[cls * DIMS + t] = v / fmaxf(norm, 1e-12f);
    }
    if (t < 4) out[4 * DIMS + t] = (float)t;         // target = arange(4)
}

extern "C" void kernel_launch(void* const* d_in, const int* in_sizes, int n_in,
                              void* d_out, int out_size, void* d_ws, size_t ws_size,
                              hipStream_t stream) {
    const float* features = (const float*)d_in[0];
    const int*   labels   = (const int*)d_in[1];
    float* out = (float*)d_out;
    float* ws  = (float*)d_ws;          // [0..1023] seg_sum, [1024..1027] counts
    const int N = in_sizes[1];          // 1,000,000 rows

    zero_ws_kernel<<<(4 * DIMS + 4 + 255) / 256, 256, 0, stream>>>(ws, 4 * DIMS + 4);

    const int blocks = (N + ROWS_PER_BLOCK - 1) / ROWS_PER_BLOCK;
    seg_sum_wmma_kernel<<<blocks, BLOCK_THREADS, 0, stream>>>(features, labels, ws, N);

    count_labels_kernel<<<1024, 256, 0, stream>>>(labels, ws + 4 * DIMS, N);

    finalize_kernel<<<1, 256, 0, stream>>>(ws, out);
}